// GATLayer_48756468744912
// MI455X (gfx1250) — compile-verified
//
#include <hip/hip_runtime.h>
#include <math.h>

typedef float v2f __attribute__((ext_vector_type(2)));
typedef float v8f __attribute__((ext_vector_type(8)));

#define DIM 256             // IN_DIM == OUT_DIM == 256
#define INV_SQRT_HD 0.125f  // 1/sqrt(HEAD_DIM=64)

// ---------------------------------------------------------------------------
// GEMM + bias with up to 3 weight sets (selected by blockIdx.z):
//   out_z[M x 256] = A[M x 256] @ W_z[256 x 256] + bias_z
// One wave computes a 32(M) x 64(N) strip with V_WMMA_F32_16X16X4_F32:
// two A fragments share each B fragment -> 8 WMMA per K-step.
// Block = 256 threads = 8 waves -> 256 rows per block; grid.y picks N strip.
// ---------------------------------------------------------------------------
__global__ __launch_bounds__(256)
void gemm_bias_wmma3(const float* __restrict__ A,
                     const float* __restrict__ Wa, const float* __restrict__ Wb,
                     const float* __restrict__ Wc,
                     const float* __restrict__ ba, const float* __restrict__ bb,
                     const float* __restrict__ bc,
                     float* __restrict__ oa, float* __restrict__ ob,
                     float* __restrict__ oc, int M)
{
    const float* __restrict__ W    = (blockIdx.z == 0) ? Wa : (blockIdx.z == 1 ? Wb : Wc);
    const float* __restrict__ bias = (blockIdx.z == 0) ? ba : (blockIdx.z == 1 ? bb : bc);
    float*       __restrict__ out  = (blockIdx.z == 0) ? oa : (blockIdx.z == 1 ? ob : oc);

    const int lane = threadIdx.x & 31;
    const int wave = threadIdx.x >> 5;
    const int half = lane >> 4;     // 0 for lanes 0-15, 1 for lanes 16-31
    const int l16  = lane & 15;
    const int row0 = blockIdx.x * 256 + wave * 32;   // two 16-row sub-tiles
    const int n0   = blockIdx.y * 64;
    if (row0 >= M) return;          // wave-uniform: EXEC stays all-ones

    // A fragment rows for this lane (clamped; partial tiles store-predicated)
    int ar0 = row0 + l16;        if (ar0 >= M) ar0 = M - 1;
    int ar1 = row0 + 16 + l16;   if (ar1 >= M) ar1 = M - 1;
    const float* __restrict__ arow0 = A + (size_t)ar0 * DIM;
    const float* __restrict__ arow1 = A + (size_t)ar1 * DIM;

    v8f c0 = {}, c1 = {}, c2 = {}, c3 = {};   // rows row0..row0+15
    v8f c4 = {}, c5 = {}, c6 = {}, c7 = {};   // rows row0+16..row0+31

#pragma unroll 2
    for (int k = 0; k < DIM; k += 4) {
        const int ka = k + 2 * half;          // this lane's K pair: ka, ka+1
        // A 16x4 fragments: contiguous float2 per lane
        v2f a0; a0.x = arow0[ka]; a0.y = arow0[ka + 1];
        v2f a1; a1.x = arow1[ka]; a1.y = arow1[ka + 1];
        // B 4x16 fragments for 4 N-tiles: VGPR0 = row ka, VGPR1 = row ka+1
        const float* __restrict__ w0 = W + (size_t)ka * DIM + n0 + l16;
        const float* __restrict__ w1 = w0 + DIM;
        v2f b0; b0.x = w0[0];  b0.y = w1[0];
        v2f b1; b1.x = w0[16]; b1.y = w1[16];
        v2f b2; b2.x = w0[32]; b2.y = w1[32];
        v2f b3; b3.x = w0[48]; b3.y = w1[48];
        // near-cache prefetch of next K-step of W (high locality -> WGP scope)
        if (k + 4 < DIM) __builtin_prefetch(w0 + 4 * DIM, 0, 3);

        c0 = __builtin_amdgcn_wmma_f32_16x16x4_f32(false, a0, false, b0,
                                                   (short)0, c0, false, false);
        c4 = __builtin_amdgcn_wmma_f32_16x16x4_f32(false, a1, false, b0,
                                                   (short)0, c4, false, false);
        c1 = __builtin_amdgcn_wmma_f32_16x16x4_f32(false, a0, false, b1,
                                                   (short)0, c1, false, false);
        c5 = __builtin_amdgcn_wmma_f32_16x16x4_f32(false, a1, false, b1,
                                                   (short)0, c5, false, false);
        c2 = __builtin_amdgcn_wmma_f32_16x16x4_f32(false, a0, false, b2,
                                                   (short)0, c2, false, false);
        c6 = __builtin_amdgcn_wmma_f32_16x16x4_f32(false, a1, false, b2,
                                                   (short)0, c6, false, false);
        c3 = __builtin_amdgcn_wmma_f32_16x16x4_f32(false, a0, false, b3,
                                                   (short)0, c3, false, false);
        c7 = __builtin_amdgcn_wmma_f32_16x16x4_f32(false, a1, false, b3,
                                                   (short)0, c7, false, false);
    }

    // Epilogue: C/D layout -> VGPR j holds M = j + 8*half, N = l16 (+tile*16)
    const float bb0 = bias[n0 + l16];
    const float bb1 = bias[n0 + 16 + l16];
    const float bb2 = bias[n0 + 32 + l16];
    const float bb3 = bias[n0 + 48 + l16];
#pragma unroll
    for (int j = 0; j < 8; ++j) {
        const int r = row0 + j + 8 * half;
        if (r < M) {
            float* __restrict__ o = out + (size_t)r * DIM + n0 + l16;
            o[0]  = c0[j] + bb0;
            o[16] = c1[j] + bb1;
            o[32] = c2[j] + bb2;
            o[48] = c3[j] + bb3;
        }
        const int r2 = row0 + 16 + j + 8 * half;
        if (r2 < M) {
            float* __restrict__ o = out + (size_t)r2 * DIM + n0 + l16;
            o[0]  = c4[j] + bb0;
            o[16] = c5[j] + bb1;
            o[32] = c6[j] + bb2;
            o[48] = c7[j] + bb3;
        }
    }
}

// ---------------------------------------------------------------------------
// Edge phase: one wave32 per edge.
// Each lane owns 8 of the 256 channels (two float4 loads), computes a partial
// dot(q[dst], k[src]), butterfly-reduces across the wave, then scatters
// attn * v[src] with global_atomic_add_f32.
// ---------------------------------------------------------------------------
__global__ __launch_bounds__(256)
void edge_attn_scatter(const float* __restrict__ q, const float* __restrict__ k,
                       const float* __restrict__ v, const int* __restrict__ ei,
                       float* __restrict__ agg, int nEdges)
{
    const int lane = threadIdx.x & 31;
    const int wave = threadIdx.x >> 5;
    const int e = blockIdx.x * 8 + wave;
    if (e >= nEdges) return;

    const int src = ei[e];
    const int dst = ei[nEdges + e];

    const float4* __restrict__ qp = (const float4*)(q + (size_t)dst * DIM);
    const float4* __restrict__ kp = (const float4*)(k + (size_t)src * DIM);
    const float4 q0 = qp[2 * lane], q1 = qp[2 * lane + 1];
    const float4 k0 = kp[2 * lane], k1 = kp[2 * lane + 1];

    float part = q0.x * k0.x + q0.y * k0.y + q0.z * k0.z + q0.w * k0.w
               + q1.x * k1.x + q1.y * k1.y + q1.z * k1.z + q1.w * k1.w;

    // wave32 butterfly reduction -> every lane holds the full dot product
#pragma unroll
    for (int off = 16; off > 0; off >>= 1)
        part += __shfl_xor(part, off, 32);

    const float attn = 1.0f / (1.0f + expf(-part * INV_SQRT_HD));

    const float4* __restrict__ vp = (const float4*)(v + (size_t)src * DIM);
    const float4 v0 = vp[2 * lane], v1 = vp[2 * lane + 1];

    float* __restrict__ o = agg + (size_t)dst * DIM + lane * 8;
    atomicAdd(o + 0, attn * v0.x);
    atomicAdd(o + 1, attn * v0.y);
    atomicAdd(o + 2, attn * v0.z);
    atomicAdd(o + 3, attn * v0.w);
    atomicAdd(o + 4, attn * v1.x);
    atomicAdd(o + 5, attn * v1.y);
    atomicAdd(o + 6, attn * v1.z);
    atomicAdd(o + 7, attn * v1.w);
}

__global__ void fill_zero(float* __restrict__ p, int n)
{
    int i = blockIdx.x * blockDim.x + threadIdx.x;
    const int stride = gridDim.x * blockDim.x;
    for (; i < n; i += stride) p[i] = 0.0f;
}

// ---------------------------------------------------------------------------
// Inputs (setup_inputs order):
//  0:x (1,20000,256) f32   1:edge_index (2,320000) i32
//  2:Wq 3:bq 4:Wk 5:bk 6:Wv 7:bv 8:Wo 9:bo        (all f32)
// d_out: (1,20000,256) f32
// Workspace: q | k | v | agg, each M*256 f32  (~82 MB total)
// ---------------------------------------------------------------------------
extern "C" void kernel_launch(void* const* d_in, const int* in_sizes, int n_in,
                              void* d_out, int out_size, void* d_ws, size_t ws_size,
                              hipStream_t stream)
{
    const float* x  = (const float*)d_in[0];
    const int*   ei = (const int*)d_in[1];
    const float* Wq = (const float*)d_in[2];
    const float* bq = (const float*)d_in[3];
    const float* Wk = (const float*)d_in[4];
    const float* bk = (const float*)d_in[5];
    const float* Wv = (const float*)d_in[6];
    const float* bv = (const float*)d_in[7];
    const float* Wo = (const float*)d_in[8];
    const float* bo = (const float*)d_in[9];
    float* out = (float*)d_out;

    const int M      = in_sizes[0] / DIM;  // 20000 nodes
    const int nEdges = in_sizes[1] / 2;    // 320000 edges
    const size_t mat = (size_t)M * DIM;

    float* ws  = (float*)d_ws;
    float* q   = ws;
    float* kk  = q + mat;
    float* vv  = kk + mat;
    float* agg = vv + mat;

    const dim3 blk(256);
    const dim3 gQKV((M + 255) / 256, 4, 3);  // fused q,k,v GEMMs
    const dim3 gO((M + 255) / 256, 4, 1);    // final @Wo GEMM

    gemm_bias_wmma3<<<gQKV, blk, 0, stream>>>(x, Wq, Wk, Wv, bq, bk, bv,
                                              q, kk, vv, M);

    fill_zero<<<1024, 256, 0, stream>>>(agg, (int)mat);

    edge_attn_scatter<<<(nEdges + 7) / 8, blk, 0, stream>>>(q, kk, vv, ei, agg, nEdges);

    gemm_bias_wmma3<<<gO, blk, 0, stream>>>(agg, Wo, Wo, Wo, bo, bo, bo,
                                            out, out, out, M);
}